// GAT_sparse_32126355374976
// MI455X (gfx1250) — compile-verified
//
#include <hip/hip_runtime.h>
#include <hip/hip_bf16.h>
#include <math.h>

// GAT, 2 layers. N=50000 nodes, E=800000 edges, IN_F=256, HID=64, HEADS=4, OUT=64.
// Dense GEMMs: fp32 WMMA (v_wmma_f32_16x16x4_f32) with async-to-LDS double-buffered
// tile staging (global_load_async_to_lds_b128 + s_wait_asynccnt) when available.
// Edge softmax/aggregate: L2-resident float4 gathers + float atomics
// (graph working set << 192MB L2).

#define NEG_SLOPE 0.2f
#define INF 256      // layer0 K
#define HID 64
#define HEADS 4
#define OUTF 64

typedef float v2f __attribute__((ext_vector_type(2)));
typedef float v8f __attribute__((ext_vector_type(8)));
typedef int   i4x __attribute__((vector_size(16)));   // matches builtin param type

#if defined(__has_builtin)
#  if __has_builtin(__builtin_amdgcn_global_load_async_to_lds_b128)
#    define ASYNC_LDS 1
#  endif
#endif
#ifndef ASYNC_LDS
#  define ASYNC_LDS 0
#endif

__device__ __forceinline__ void stage16(const float* __restrict__ g,
                                        float* __restrict__ l) {
#if ASYNC_LDS
  __builtin_amdgcn_global_load_async_to_lds_b128(
      (__attribute__((address_space(1))) i4x*)g,
      (__attribute__((address_space(3))) i4x*)l, 0, 0);
#else
  *(float4*)l = *(const float4*)g;
#endif
}

__device__ __forceinline__ void wait_async_zero() {
#if ASYNC_LDS
#  if __has_builtin(__builtin_amdgcn_s_wait_asynccnt)
  __builtin_amdgcn_s_wait_asynccnt(0);
#  else
  asm volatile("s_wait_asynccnt 0" ::: "memory");
#  endif
#endif
}

// ---------------------------------------------------------------------------
// C[M,N] = A[M,K] * W[N,K]^T  using V_WMMA_F32_16X16X4_F32.
// Block tile 64x64, K in chunks of 32, double-buffered LDS filled by async
// b128 copies. 256 threads = 8 waves; wave owns a 32(M)x16(N) sub-tile.
// Both tiles stored row-major in k: As[m][k], Bs[n][k] (stride 36 -> rows
// 16B aligned, fragment reads bank-conflict free).
// ---------------------------------------------------------------------------
__global__ __launch_bounds__(256)
void gemm_wmma_f32(const float* __restrict__ A, const float* __restrict__ W,
                   float* __restrict__ C, int M, int N, int K) {
  __shared__ float As[2][64][36];
  __shared__ float Bs[2][64][36];

  const int tid  = threadIdx.x;
  const int lane = tid & 31;
  const int wave = tid >> 5;
  const int half = lane >> 4;    // 0: lanes 0-15, 1: lanes 16-31
  const int lrow = lane & 15;
  const int m0   = blockIdx.x * 64;
  const int n0   = blockIdx.y * 64;
  const int ncol  = (wave & 3) * 16;   // wave's col tile within block
  const int mbase = (wave >> 2) * 32;  // wave's 32-row group within block

  v8f acc0 = {}; v8f acc1 = {};

  auto stage_both = [&](int k0, int buf) {
    // A tile: rows m0..m0+63 (clamped; clamped rows only feed masked C rows)
    for (int i = tid; i < 512; i += 256) {
      int r = i >> 3, cc = (i & 7) * 4;
      int gr = m0 + r; if (gr >= M) gr = M - 1;
      stage16(&A[(size_t)gr * K + k0 + cc], &As[buf][r][cc]);
    }
    // B tile: W rows n0..n0+63 (clamped), stored un-transposed: Bs[n][k]
    for (int i = tid; i < 512; i += 256) {
      int n = i >> 3, cc = (i & 7) * 4;
      int gn = n0 + n; if (gn >= N) gn = N - 1;
      stage16(&W[(size_t)gn * K + k0 + cc], &Bs[buf][n][cc]);
    }
  };

  stage_both(0, 0);
  wait_async_zero();
  __syncthreads();

  int buf = 0;
  for (int k0 = 0; k0 < K; k0 += 32) {
    if (k0 + 32 < K) stage_both(k0 + 32, buf ^ 1);   // overlap DMA with WMMA

    // 16x16x4 f32 WMMA fragments:
    //  A: lanes 0-15 {v0=K0,v1=K1}, lanes 16-31 {v0=K2,v1=K3}, M=lane%16
    //  B: lanes 0-15 {v0=K0,v1=K1}, lanes 16-31 {v0=K2,v1=K3}, N=lane%16
    #pragma unroll
    for (int kk = 0; kk < 32; kk += 4) {
      const int kb = kk + 2 * half;
      v2f b, a0, a1;
      b.x  = Bs[buf][ncol + lrow][kb + 0];
      b.y  = Bs[buf][ncol + lrow][kb + 1];
      a0.x = As[buf][mbase + lrow][kb + 0];
      a0.y = As[buf][mbase + lrow][kb + 1];
      a1.x = As[buf][mbase + 16 + lrow][kb + 0];
      a1.y = As[buf][mbase + 16 + lrow][kb + 1];
      acc0 = __builtin_amdgcn_wmma_f32_16x16x4_f32(false, a0, false, b,
                                                   (short)0, acc0, false, false);
      acc1 = __builtin_amdgcn_wmma_f32_16x16x4_f32(false, a1, false, b,
                                                   (short)0, acc1, false, false);
    }

    wait_async_zero();   // this wave's prefetch landed
    __syncthreads();     // everyone done reading buf / writing buf^1
    buf ^= 1;
  }

  // D layout: VGPR r -> M = r + 8*half, N = lane%16.
  const int gcol = n0 + ncol + lrow;
  if (gcol < N) {
    #pragma unroll
    for (int r = 0; r < 8; ++r) {
      int g0 = m0 + mbase + r + 8 * half;
      if (g0 < M) C[(size_t)g0 * N + gcol] = acc0[r];
      int g1 = m0 + mbase + 16 + r + 8 * half;
      if (g1 < M) C[(size_t)g1 * N + gcol] = acc1[r];
    }
  }
}

// ---------------------------------------------------------------------------
// el/er per (node, head): el = <h[n,h,:], al[h,:]>, er = <h[n,h,:], ar[h,:]>
// dim is a multiple of 4.
// ---------------------------------------------------------------------------
__global__ void attn_coeff(const float* __restrict__ h, const float* __restrict__ al,
                           const float* __restrict__ ar, float* __restrict__ el,
                           float* __restrict__ er, int n_nodes, int heads, int dim) {
  int idx = blockIdx.x * blockDim.x + threadIdx.x;   // node*heads + head
  if (idx >= n_nodes * heads) return;
  int head = idx % heads;
  const float4* hp  = (const float4*)(h + (size_t)idx * dim);
  const float4* alp = (const float4*)(al + (size_t)head * dim);
  const float4* arp = (const float4*)(ar + (size_t)head * dim);
  float sl = 0.f, sr = 0.f;
  for (int d = 0; d < (dim >> 2); ++d) {
    float4 v = hp[d], a = alp[d], b = arp[d];
    sl += v.x * a.x + v.y * a.y + v.z * a.z + v.w * a.w;
    sr += v.x * b.x + v.y * b.y + v.z * b.z + v.w * b.w;
  }
  el[idx] = sl; er[idx] = sr;
}

// ---------------------------------------------------------------------------
__global__ void fill_f32(float* __restrict__ p, float v, long n) {
  long i = (long)blockIdx.x * blockDim.x + threadIdx.x;
  if (i < n) p[i] = v;
}
__global__ void fill_bias(float* __restrict__ p, const float* __restrict__ b,
                          int cols, long n) {
  long i = (long)blockIdx.x * blockDim.x + threadIdx.x;
  if (i < n) p[i] = b[i % cols];
}

__device__ inline void atomicMaxF(float* addr, float v) {
  // valid for buffers initialized to -inf
  if (v >= 0.f) atomicMax((int*)addr, __float_as_int(v));
  else          atomicMin((unsigned int*)addr, (unsigned int)__float_as_int(v));
}

// pass 1: e = leaky_relu(el[src]+er[dst]); segment max into m[dst]
__global__ void edge_logits_max(const int* __restrict__ src, const int* __restrict__ dst,
                                const float* __restrict__ el, const float* __restrict__ er,
                                float* __restrict__ e, float* __restrict__ m,
                                int n_edges, int heads) {
  int i = blockIdx.x * blockDim.x + threadIdx.x;   // edge*heads + head
  if (i >= n_edges * heads) return;
  int eid = i / heads, h = i % heads;
  int s = src[eid], d = dst[eid];
  float v = el[s * heads + h] + er[d * heads + h];
  v = (v >= 0.f) ? v : NEG_SLOPE * v;
  e[i] = v;
  atomicMaxF(&m[d * heads + h], v);
}

// m = isfinite(m) ? m : 0
__global__ void fix_max(float* __restrict__ m, long n) {
  long i = (long)blockIdx.x * blockDim.x + threadIdx.x;
  if (i < n) {
    unsigned u = __float_as_uint(m[i]);
    if ((u & 0x7f800000u) == 0x7f800000u) m[i] = 0.f;
  }
}

// pass 2: ee = exp(e - m[dst]); segment sum into s[dst]
__global__ void edge_exp_sum(const int* __restrict__ dst, float* __restrict__ e,
                             const float* __restrict__ m, float* __restrict__ s,
                             int n_edges, int heads) {
  int i = blockIdx.x * blockDim.x + threadIdx.x;
  if (i >= n_edges * heads) return;
  int eid = i / heads, h = i % heads;
  int d = dst[eid];
  float ee = __expf(e[i] - m[d * heads + h]);
  e[i] = ee;
  atomicAdd(&s[d * heads + h], ee);
}

// pass 2b: alpha = ee / s[dst]  (once per edge-head, keeps divides out of pass 3)
__global__ void edge_alpha(const int* __restrict__ dst, float* __restrict__ e,
                           const float* __restrict__ s, int n_edges, int heads) {
  int i = blockIdx.x * blockDim.x + threadIdx.x;
  if (i >= n_edges * heads) return;
  int eid = i / heads, h = i % heads;
  e[i] = e[i] / s[dst[eid] * heads + h];
}

// pass 3: one wave per edge; acc[dst] += h[src] * alpha; dim == 64.
// float4 gathers (L2-resident) + global_atomic_add_f32 scatters.
__global__ __launch_bounds__(256)
void edge_aggregate(const int* __restrict__ src, const int* __restrict__ dst,
                    const float* __restrict__ h, const float* __restrict__ e,
                    float* __restrict__ acc, int n_edges, int heads) {
  int wv   = (int)(((long)blockIdx.x * blockDim.x + threadIdx.x) >> 5);
  int lane = threadIdx.x & 31;
  if (wv >= n_edges) return;
  int sn = src[wv], dn = dst[wv];
  const int hd  = heads << 6;          // heads*64
  const int nch = hd >> 2;             // 16B chunks (16 per head)
  const float4* hrow = (const float4*)(h + (size_t)sn * hd);
  float* arow = acc + (size_t)dn * hd;
  for (int c = lane; c < nch; c += 32) {
    float alpha = e[(size_t)wv * heads + (c >> 4)];   // dim=64 -> 16 chunks/head
    float4 v = hrow[c];
    float* ap = arow + 4 * c;
    atomicAdd(ap + 0, v.x * alpha);
    atomicAdd(ap + 1, v.y * alpha);
    atomicAdd(ap + 2, v.z * alpha);
    atomicAdd(ap + 3, v.w * alpha);
  }
}

// h1 = relu(acc + b)  (in place)
__global__ void relu_bias(float* __restrict__ acc, const float* __restrict__ b,
                          int cols, long n) {
  long i = (long)blockIdx.x * blockDim.x + threadIdx.x;
  if (i < n) {
    float v = acc[i] + b[i % cols];
    acc[i] = v > 0.f ? v : 0.f;
  }
}

// ---------------------------------------------------------------------------
extern "C" void kernel_launch(void* const* d_in, const int* in_sizes, int n_in,
                              void* d_out, int out_size, void* d_ws, size_t ws_size,
                              hipStream_t stream) {
  const float* x   = (const float*)d_in[0];
  const int*   src = (const int*)  d_in[1];
  const int*   dst = (const int*)  d_in[2];
  const float* W0  = (const float*)d_in[3];
  const float* al0 = (const float*)d_in[4];
  const float* ar0 = (const float*)d_in[5];
  const float* b0  = (const float*)d_in[6];
  const float* W1  = (const float*)d_in[7];
  const float* al1 = (const float*)d_in[8];
  const float* ar1 = (const float*)d_in[9];
  const float* b1  = (const float*)d_in[10];
  float* out = (float*)d_out;

  const int nN = in_sizes[0] / INF;   // 50000
  const int nE = in_sizes[1];         // 800000
  const int HD0 = HEADS * HID;        // 256

  // workspace layout (floats); ~123 MB total
  float* w   = (float*)d_ws;
  float* h0  = w;  w += (size_t)nN * HD0;     // layer0 GEMM out, reused as layer1 GEMM out
  float* acc0= w;  w += (size_t)nN * HD0;     // layer0 aggregate -> h1 after relu+bias
  float* el0 = w;  w += (size_t)nN * HEADS;
  float* er0 = w;  w += (size_t)nN * HEADS;
  float* m0  = w;  w += (size_t)nN * HEADS;
  float* s0  = w;  w += (size_t)nN * HEADS;
  float* e0  = w;  w += (size_t)nE * HEADS;
  float* el1 = w;  w += nN;
  float* er1 = w;  w += nN;
  float* m1  = w;  w += nN;
  float* s1  = w;  w += nN;
  float* e1  = w;  w += nE;

  const float NEG_INF = __builtin_huge_valf() * -1.0f;
  auto blk = [](long n, int t) { return (unsigned)((n + t - 1) / t); };

  // ---------------- layer 0 ----------------
  {
    dim3 g(blk(nN, 64), HD0 / 64);
    gemm_wmma_f32<<<g, 256, 0, stream>>>(x, W0, h0, nN, HD0, INF);
  }
  attn_coeff<<<blk((long)nN * HEADS, 256), 256, 0, stream>>>(h0, al0, ar0, el0, er0, nN, HEADS, HID);

  fill_f32<<<blk((long)nN * HEADS, 256), 256, 0, stream>>>(m0, NEG_INF, (long)nN * HEADS);
  fill_f32<<<blk((long)nN * HEADS, 256), 256, 0, stream>>>(s0, 0.f, (long)nN * HEADS);
  fill_f32<<<blk((long)nN * HD0, 256), 256, 0, stream>>>(acc0, 0.f, (long)nN * HD0);

  edge_logits_max<<<blk((long)nE * HEADS, 256), 256, 0, stream>>>(src, dst, el0, er0, e0, m0, nE, HEADS);
  fix_max<<<blk((long)nN * HEADS, 256), 256, 0, stream>>>(m0, (long)nN * HEADS);
  edge_exp_sum<<<blk((long)nE * HEADS, 256), 256, 0, stream>>>(dst, e0, m0, s0, nE, HEADS);
  edge_alpha<<<blk((long)nE * HEADS, 256), 256, 0, stream>>>(dst, e0, s0, nE, HEADS);
  edge_aggregate<<<blk((long)nE * 32, 256), 256, 0, stream>>>(src, dst, h0, e0, acc0, nE, HEADS);

  relu_bias<<<blk((long)nN * HD0, 256), 256, 0, stream>>>(acc0, b0, HD0, (long)nN * HD0);  // acc0 -> h1

  // ---------------- layer 1 ----------------
  {
    dim3 g(blk(nN, 64), 1);  // N = 64
    gemm_wmma_f32<<<g, 256, 0, stream>>>(acc0, W1, h0, nN, OUTF, HD0);  // h0 reused as layer1 features
  }
  attn_coeff<<<blk(nN, 256), 256, 0, stream>>>(h0, al1, ar1, el1, er1, nN, 1, OUTF);

  fill_f32<<<blk(nN, 256), 256, 0, stream>>>(m1, NEG_INF, nN);
  fill_f32<<<blk(nN, 256), 256, 0, stream>>>(s1, 0.f, nN);
  fill_bias<<<blk((long)nN * OUTF, 256), 256, 0, stream>>>(out, b1, OUTF, (long)nN * OUTF);

  edge_logits_max<<<blk(nE, 256), 256, 0, stream>>>(src, dst, el1, er1, e1, m1, nE, 1);
  fix_max<<<blk(nN, 256), 256, 0, stream>>>(m1, nN);
  edge_exp_sum<<<blk(nE, 256), 256, 0, stream>>>(dst, e1, m1, s1, nE, 1);
  edge_alpha<<<blk(nE, 256), 256, 0, stream>>>(dst, e1, s1, nE, 1);
  edge_aggregate<<<blk((long)nE * 32, 256), 256, 0, stream>>>(src, dst, h0, e1, out, nE, 1);
}